// RegHD_AR_48996986913338
// MI455X (gfx1250) — compile-verified
//
#include <hip/hip_runtime.h>
#include <hip/hip_bf16.h>

#define SIZE    1024
#define D       16384
#define ROWS    32          // d-rows per block
#define TSTRIDE 1028        // padded LDS row stride in floats (16B-aligned, conflict-free)

typedef __attribute__((ext_vector_type(2))) float v2f;
typedef __attribute__((ext_vector_type(8))) float v8f;

__global__ __launch_bounds__(256) void reghd_encode_kernel(
    const float* __restrict__ x,
    const float* __restrict__ weight,
    const float* __restrict__ bias,
    const float* __restrict__ alpha,
    float* __restrict__ encq,      // (D,)
    float* __restrict__ hvs)       // (SIZE, D) row-major = enc.T
{
    __shared__ float xs[SIZE];
    __shared__ float alds[SIZE];
    __shared__ float tile[ROWS * TSTRIDE];   // enc tile, padded rows
    __shared__ float rowsum[ROWS];

    const int t    = threadIdx.x;     // 0..255
    const int lane = t & 31;
    const int wave = t >> 5;          // 0..7
    const int d0   = blockIdx.x * ROWS;

    // ---- stage x and alpha into LDS (256 threads x float4 = 1024 floats each) ----
    ((float4*)xs)[t]   = ((const float4*)x)[t];
    ((float4*)alds)[t] = ((const float4*)alpha)[t];
    if (t < ROWS) rowsum[t] = 0.0f;
    __syncthreads();

    // ---- compute enc for 4 rows per wave, coalesced B128 loads along s ----
    #pragma unroll
    for (int rr = 0; rr < 4; ++rr) {
        const int r = wave * 4 + rr;
        const float4* __restrict__ wrow = (const float4*)(weight + (size_t)(d0 + r) * SIZE);
        const float4* __restrict__ brow = (const float4*)(bias   + (size_t)(d0 + r) * SIZE);
        float4* __restrict__ trow4 = (float4*)(tile + r * TSTRIDE);
        #pragma unroll
        for (int it = 0; it < 8; ++it) {
            const int sv = it * 32 + lane;          // float4 index, s = sv*4
            float4 w  = wrow[sv];
            float4 b  = brow[sv];
            float4 xv = ((const float4*)xs)[sv];
            float4 e;
            float p;
            p   = xv.x * w.x;  e.x = __cosf(p + b.x) * __sinf(p);
            p   = xv.y * w.y;  e.y = __cosf(p + b.y) * __sinf(p);
            p   = xv.z * w.z;  e.z = __cosf(p + b.z) * __sinf(p);
            p   = xv.w * w.w;  e.w = __cosf(p + b.w) * __sinf(p);
            trow4[sv] = e;
        }
    }
    __syncthreads();

    // ---- WMMA row-sum: C += A(16x4) * ones(4x16), exact f32 accumulation ----
    // waves 0-3: rows 0-15, k-slice (wave&3)*256; waves 4-7: rows 16-31
    {
        const int g    = wave >> 2;                 // row group
        const int q    = wave & 3;                  // k slice
        const int arow = (lane & 15) + g * 16;      // A-matrix row for this lane
        const int koff = (lane & 16) ? 2 : 0;       // upper half-wave holds K+2,K+3
        const float* __restrict__ trow = tile + arow * TSTRIDE;

        v8f c = {};
        v2f bones; bones[0] = 1.0f; bones[1] = 1.0f;

        #pragma unroll 8
        for (int kc = 0; kc < 64; ++kc) {
            const int k = q * 256 + kc * 4 + koff;
            v2f a;
            a[0] = trow[k]     * alds[k];           // enc * alpha
            a[1] = trow[k + 1] * alds[k + 1];
            c = __builtin_amdgcn_wmma_f32_16x16x4_f32(
                    /*neg_a=*/false, a, /*neg_b=*/false, bones,
                    /*c_mod=*/(short)0, c, /*reuse_a=*/false, /*reuse_b=*/false);
        }

        // D layout: vgpr v, lanes0-15 -> M=v (N=lane), lanes16-31 -> M=v+8.
        // Column N=0 (lane 0) / N=0 of upper half (lane 16) carry full row sums.
        if ((lane & 15) == 0) {
            const int base = g * 16 + ((lane & 16) ? 8 : 0);
            #pragma unroll
            for (int i = 0; i < 8; ++i)
                atomicAdd(&rowsum[base + i], c[i]);   // ds_add_f32 across k-slices
        }
    }
    __syncthreads();

    // ---- enc_q = hard_quantize(rowsum) ----
    if (t < ROWS)
        encq[d0 + t] = rowsum[t] > 0.0f ? 1.0f : -1.0f;

    // ---- hvs transpose write-out: each s-row gets 32 contiguous floats (128B line) ----
    #pragma unroll
    for (int j = 0; j < 4; ++j) {
        const int s = t + j * 256;
        float4* __restrict__ dst = (float4*)(hvs + (size_t)s * D + d0);
        #pragma unroll
        for (int q = 0; q < 8; ++q) {
            float4 v;
            v.x = tile[(q * 4 + 0) * TSTRIDE + s];
            v.y = tile[(q * 4 + 1) * TSTRIDE + s];
            v.z = tile[(q * 4 + 2) * TSTRIDE + s];
            v.w = tile[(q * 4 + 3) * TSTRIDE + s];
            dst[q] = v;
        }
    }
}

__global__ __launch_bounds__(256) void reghd_dot_kernel(
    const float* __restrict__ encq,
    const float* __restrict__ M,
    float* __restrict__ out)
{
    __shared__ float red[256];
    float acc = 0.0f;
    for (int i = threadIdx.x; i < D; i += 256)
        acc += encq[i] * M[i];
    red[threadIdx.x] = acc;
    __syncthreads();
    #pragma unroll
    for (int ofs = 128; ofs > 0; ofs >>= 1) {
        if (threadIdx.x < ofs) red[threadIdx.x] += red[threadIdx.x + ofs];
        __syncthreads();
    }
    if (threadIdx.x == 0) out[0] = red[0];
}

extern "C" void kernel_launch(void* const* d_in, const int* in_sizes, int n_in,
                              void* d_out, int out_size, void* d_ws, size_t ws_size,
                              hipStream_t stream) {
    // inputs: x, ts, weight, bias, alpha, M
    const float* x      = (const float*)d_in[0];
    const float* weight = (const float*)d_in[2];
    const float* bias   = (const float*)d_in[3];
    const float* alpha  = (const float*)d_in[4];
    const float* M      = (const float*)d_in[5];

    float* out       = (float*)d_out;
    float* out_model = out;            // scalar
    float* out_encq  = out + 1;        // (D,)
    float* out_hvs   = out + 1 + D;    // (SIZE, D)

    reghd_encode_kernel<<<D / ROWS, 256, 0, stream>>>(x, weight, bias, alpha,
                                                      out_encq, out_hvs);
    reghd_dot_kernel<<<1, 256, 0, stream>>>(out_encq, M, out_model);
}